// SiameseKPConv_69715909148860
// MI455X (gfx1250) — compile-verified
//
#include <hip/hip_runtime.h>

#define NPTS   16384
#define NBATCH 8
#define NPB    (NPTS / NBATCH)
#define KNBR   16
#define KP     15
#define SIGMA  0.1f

typedef float v2f __attribute__((ext_vector_type(2)));
typedef float v8f __attribute__((ext_vector_type(8)));

// ---------------- kNN (self, k=16) : one thread per query, batch-local scan --------
__global__ void knn16_kernel(const float* __restrict__ pos, int* __restrict__ out) {
  int n = blockIdx.x * blockDim.x + threadIdx.x;
  if (n >= NPTS) return;
  int base = (n / NPB) * NPB;
  float qx = pos[3*n], qy = pos[3*n+1], qz = pos[3*n+2];
  float bd[KNBR]; int bi[KNBR];
#pragma unroll
  for (int j = 0; j < KNBR; ++j) { bd[j] = 3.4e38f; bi[j] = base; }
  for (int m = 0; m < NPB; ++m) {
    int s = base + m;
    float dx = qx - pos[3*s], dy = qy - pos[3*s+1], dz = qz - pos[3*s+2];
    float d2 = dx*dx + dy*dy + dz*dz;
    if (d2 < bd[KNBR-1]) {
      int j = KNBR - 1;
      while (j > 0 && bd[j-1] > d2) { bd[j] = bd[j-1]; bi[j] = bi[j-1]; --j; }
      bd[j] = d2; bi[j] = s;
    }
  }
  for (int j = 0; j < KNBR; ++j) out[n*KNBR + j] = bi[j];
}

// ---------------- nearest neighbor (cross, k=1) -----------------------------------
__global__ void nn1_kernel(const float* __restrict__ q, const float* __restrict__ s,
                           int* __restrict__ out) {
  int n = blockIdx.x * blockDim.x + threadIdx.x;
  if (n >= NPTS) return;
  int base = (n / NPB) * NPB;
  float qx = q[3*n], qy = q[3*n+1], qz = q[3*n+2];
  float best = 3.4e38f; int bidx = base;
  for (int m = 0; m < NPB; ++m) {
    int ss = base + m;
    float dx = qx - s[3*ss], dy = qy - s[3*ss+1], dz = qz - s[3*ss+2];
    float d2 = dx*dx + dy*dy + dz*dz;
    if (d2 < best) { best = d2; bidx = ss; }
  }
  out[n] = bidx;
}

// ---------------- kernel-point influence weights, fixed across all layers ---------
__global__ void winf_kernel(const float* __restrict__ pos, const int* __restrict__ nidx,
                            const float* __restrict__ kp, float* __restrict__ winf) {
  int t = blockIdx.x * blockDim.x + threadIdx.x;
  if (t >= NPTS * KNBR) return;
  int n = t >> 4;
  int nb = nidx[t];
  float rx = pos[3*nb]   - pos[3*n];
  float ry = pos[3*nb+1] - pos[3*n+1];
  float rz = pos[3*nb+2] - pos[3*n+2];
  for (int p = 0; p < KP; ++p) {
    float dx = rx - kp[3*p], dy = ry - kp[3*p+1], dz = rz - kp[3*p+2];
    float d = sqrtf(dx*dx + dy*dy + dz*dz + 1e-12f);
    winf[(size_t)t*KP + p] = fmaxf(0.0f, 1.0f - d * (1.0f / SIGMA));
  }
}

// ---------------- initial features [pos | 1] --------------------------------------
__global__ void initfeat_kernel(const float* __restrict__ pos, const float* __restrict__ feat,
                                float* __restrict__ x) {
  int n = blockIdx.x * blockDim.x + threadIdx.x;
  if (n >= NPTS) return;
  x[n*4+0] = pos[3*n]; x[n*4+1] = pos[3*n+1]; x[n*4+2] = pos[3*n+2]; x[n*4+3] = feat[n];
}

// ---------------- fused KPConv: gather + agg + WMMA fp32 GEMM ---------------------
// 128 threads = 4 waves per 16-point tile. All waves share one staged
// influence/gather/agg pipeline in LDS; wave w owns output channels
// [w*64, w*64+64) (4 WMMA otiles), so one block covers Cout <= 256.
// A operand (16x4 f32): VGPR v, lane-half h -> K = 2h+v, lanes 0..15 give M.
// B operand (4x16 f32): VGPR v, lane-half h -> K = 2h+v, lanes 0..15 give N.
__global__ __launch_bounds__(128) void kpconv_wmma_kernel(
    const float* __restrict__ xin, const int* __restrict__ nidx,
    const float* __restrict__ winf, const float* __restrict__ Wm,
    float* __restrict__ out, int Cin, int Cout) {
  __shared__ float sw[16 * KNBR * KP];       // influence tile [m][k][p]
  __shared__ int   sidx[16 * KNBR];          // neighbor indices [m][k]
  __shared__ float sXg[16 * KNBR * 17];      // gathered neighbor features [(m,k)][cc]
  __shared__ float sA[16 * 17];              // agg tile, [k][m] (padded)
  int tid = threadIdx.x;
  int wv = tid >> 5;                         // wave id 0..3 -> 64-channel group
  int h = (tid >> 4) & 1, ln = tid & 15;
  int ntile = blockIdx.x;
  int obase0 = wv * 64;

  for (int t = tid; t < 16 * KNBR * KP; t += 128)
    sw[t] = winf[(size_t)ntile * (16 * KNBR * KP) + t];
  for (int t = tid; t < 16 * KNBR; t += 128)
    sidx[t] = nidx[(size_t)ntile * (16 * KNBR) + t];
  __syncthreads();

  v8f acc[4] = {};

  for (int cbase = 0; cbase < Cin; cbase += 16) {
    int cw = Cin - cbase; if (cw > 16) cw = 16;
    int c4 = cw >> 2;                        // Cin is always a multiple of 4
    // stage gathered neighbor features once per c-chunk; reused by all 15 kernel pts
    for (int r = tid; r < 16 * KNBR; r += 128) {
      const float* src = xin + (size_t)sidx[r] * Cin + cbase;
      const float4* s4 = (const float4*)src; // 16B-aligned: Cin%4==0, cbase%16==0
      float vals[16];
#pragma unroll
      for (int q = 0; q < 4; ++q) {
        float4 t = (q < c4) ? s4[q] : float4{0.0f, 0.0f, 0.0f, 0.0f};
        vals[4*q+0] = t.x; vals[4*q+1] = t.y; vals[4*q+2] = t.z; vals[4*q+3] = t.w;
      }
#pragma unroll
      for (int cc = 0; cc < 16; ++cc)
        sXg[r * 17 + cc] = vals[cc];
      if (cbase + 16 < Cin)                  // overlap next chunk fetch with 15-p compute
        __builtin_prefetch(src + 16, 0, 0);
    }
    __syncthreads();
    for (int p = 0; p < KP; ++p) {
      // agg[m][cc] = sum_k w[m,k,p] * x[nidx[m,k], cbase+cc]  (built by all 4 waves)
      for (int e = tid; e < 256; e += 128) {
        int m = e & 15, cc = e >> 4;
        float v = 0.0f;
#pragma unroll
        for (int k = 0; k < KNBR; ++k)
          v += sw[(m * KNBR + k) * KP + p] * sXg[(m * KNBR + k) * 17 + cc];
        sA[cc * 17 + m] = v;
      }
      __syncthreads();
      const float* wb = Wm + (size_t)p * Cin * Cout;
#pragma unroll
      for (int k0 = 0; k0 < 16; k0 += 4) {
        v2f a;
        a.x = sA[(k0 + h*2 + 0) * 17 + ln];
        a.y = sA[(k0 + h*2 + 1) * 17 + ln];
        int kA = cbase + k0 + h*2;
#pragma unroll
        for (int ot = 0; ot < 4; ++ot) {
          int obase = obase0 + ot * 16;
          if (obase < Cout) {                // wave-uniform branch: EXEC stays all-1s
            v2f b;
            b.x = (kA     < Cin) ? wb[(size_t)(kA    ) * Cout + obase + ln] : 0.0f;
            b.y = (kA + 1 < Cin) ? wb[(size_t)(kA + 1) * Cout + obase + ln] : 0.0f;
            acc[ot] = __builtin_amdgcn_wmma_f32_16x16x4_f32(
                false, a, false, b, (short)0, acc[ot], false, false);
          }
        }
      }
      __syncthreads();                       // before next p overwrites sA
    }
  }
  // C/D layout: VGPR r, lane-half h -> row = r + 8h, lanes give column
#pragma unroll
  for (int ot = 0; ot < 4; ++ot) {
    int obase = obase0 + ot * 16;
    if (obase < Cout) {
#pragma unroll
      for (int r = 0; r < 8; ++r)
        out[(size_t)(ntile * 16 + r + h * 8) * Cout + obase + ln] = acc[ot][r];
    }
  }
}

// ---------------- plain WMMA GEMM for the FC1 layer -------------------------------
__global__ __launch_bounds__(32) void gemm_wmma_kernel(
    const float* __restrict__ X, const float* __restrict__ W,
    float* __restrict__ out, int Cin, int Cout) {
  int lane = threadIdx.x, h = lane >> 4, ln = lane & 15;
  int ntile = blockIdx.x, obase0 = blockIdx.y * 64;
  v8f acc[4] = {};
  for (int k0 = 0; k0 < Cin; k0 += 4) {
    int kA = k0 + h*2;
    v2f a;
    a.x = X[(size_t)(ntile * 16 + ln) * Cin + kA];
    a.y = X[(size_t)(ntile * 16 + ln) * Cin + kA + 1];
#pragma unroll
    for (int ot = 0; ot < 4; ++ot) {
      int obase = obase0 + ot * 16;
      if (obase < Cout) {
        v2f b;
        b.x = W[(size_t)kA * Cout + obase + ln];
        b.y = W[(size_t)(kA + 1) * Cout + obase + ln];
        acc[ot] = __builtin_amdgcn_wmma_f32_16x16x4_f32(
            false, a, false, b, (short)0, acc[ot], false, false);
      }
    }
  }
#pragma unroll
  for (int ot = 0; ot < 4; ++ot) {
    int obase = obase0 + ot * 16;
    if (obase < Cout) {
#pragma unroll
      for (int r = 0; r < 8; ++r)
        out[(size_t)(ntile * 16 + r + h * 8) * Cout + obase + ln] = acc[ot][r];
    }
  }
}

// ---------------- batch-norm stats (biased var, matches jnp.var) ------------------
__global__ void bn_stats_kernel(const float* __restrict__ x, float* __restrict__ stats, int C) {
  __shared__ float ss[256], sq[256];
  int c = blockIdx.x, tid = threadIdx.x;
  float s = 0.0f, q = 0.0f;
  for (int i = tid; i < NPTS; i += 256) {
    float v = x[(size_t)i * C + c];
    s += v; q += v * v;
  }
  ss[tid] = s; sq[tid] = q;
  __syncthreads();
  for (int w = 128; w > 0; w >>= 1) {
    if (tid < w) { ss[tid] += ss[tid + w]; sq[tid] += sq[tid + w]; }
    __syncthreads();
  }
  if (tid == 0) {
    float m = ss[0] * (1.0f / NPTS);
    stats[c] = m;
    stats[C + c] = sq[0] * (1.0f / NPTS) - m * m;
  }
}

__global__ void bn_apply_kernel(float* __restrict__ x, const float* __restrict__ stats,
                                const float* __restrict__ g, const float* __restrict__ b,
                                float slope, int C) {
  int idx = blockIdx.x * blockDim.x + threadIdx.x;
  if (idx >= NPTS * C) return;
  int c = idx % C;
  float y = (x[idx] - stats[c]) * rsqrtf(stats[C + c] + 1e-5f) * g[c] + b[c];
  x[idx] = (y < 0.0f) ? slope * y : y;
}

// ---------------- siamese diff, skip concat, final FC -----------------------------
__global__ void diff_kernel(const float* __restrict__ x2, const float* __restrict__ x1,
                            const int* __restrict__ nn21, float* __restrict__ d, int C) {
  int idx = blockIdx.x * blockDim.x + threadIdx.x;
  if (idx >= NPTS * C) return;
  int n = idx / C, c = idx % C;
  d[idx] = x2[idx] - x1[(size_t)nn21[n] * C + c];
}

__global__ void concat_kernel(const float* __restrict__ a, const float* __restrict__ b,
                              float* __restrict__ o, int Ca, int Cb) {
  int C = Ca + Cb;
  int idx = blockIdx.x * blockDim.x + threadIdx.x;
  if (idx >= NPTS * C) return;
  int n = idx / C, c = idx % C;
  o[idx] = (c < Ca) ? a[(size_t)n * Ca + c] : b[(size_t)n * Cb + (c - Ca)];
}

__global__ void fc2_kernel(const float* __restrict__ hgt, const float* __restrict__ W2,
                           float* __restrict__ out) {
  int n = blockIdx.x * blockDim.x + threadIdx.x;
  if (n >= NPTS) return;
  float a0 = 0.0f, a1 = 0.0f;
  for (int c = 0; c < 64; ++c) {
    float hv = hgt[(size_t)n * 64 + c];
    a0 += hv * W2[c * 2];
    a1 += hv * W2[c * 2 + 1];
  }
  out[n * 2] = a0; out[n * 2 + 1] = a1;
}

// ==================================================================================
extern "C" void kernel_launch(void* const* d_in, const int* in_sizes, int n_in,
                              void* d_out, int out_size, void* d_ws, size_t ws_size,
                              hipStream_t stream) {
  (void)n_in; (void)out_size; (void)ws_size;
  const float *pos1, *feat1, *pos2, *feat2, *kp;
  const float *dW[4], *dg[4], *db[4], *uW[3], *ug[3], *ub[3], *W1, *fg, *fb, *W2;

  if (in_sizes[0] == 1) {
    // jax sorted-leaf order: B,batch1,batch2,feat1,feat2,k, down[{W,b,g}x4], fc{W1,W2,b,g}, kp, up[{W,b,g}x3], pos1,pos2
    feat1 = (const float*)d_in[3]; feat2 = (const float*)d_in[4];
    int i = 6;
    for (int blk = 0; blk < 4; ++blk) { dW[blk] = (const float*)d_in[i]; db[blk] = (const float*)d_in[i+1]; dg[blk] = (const float*)d_in[i+2]; i += 3; }
    W1 = (const float*)d_in[i]; W2 = (const float*)d_in[i+1]; fb = (const float*)d_in[i+2]; fg = (const float*)d_in[i+3]; i += 4;
    kp = (const float*)d_in[i]; ++i;
    for (int blk = 0; blk < 3; ++blk) { uW[blk] = (const float*)d_in[i]; ub[blk] = (const float*)d_in[i+1]; ug[blk] = (const float*)d_in[i+2]; i += 3; }
    pos1 = (const float*)d_in[i]; pos2 = (const float*)d_in[i+1];
  } else {
    // insertion order: pos1,feat1,pos2,feat2, kp, down[{W,g,b}x4], up[{W,g,b}x3], fc{W1,g,b,W2}, batch1,batch2,B,k
    pos1 = (const float*)d_in[0]; feat1 = (const float*)d_in[1];
    pos2 = (const float*)d_in[2]; feat2 = (const float*)d_in[3];
    kp = (const float*)d_in[4];
    int i = 5;
    for (int blk = 0; blk < 4; ++blk) { dW[blk] = (const float*)d_in[i]; dg[blk] = (const float*)d_in[i+1]; db[blk] = (const float*)d_in[i+2]; i += 3; }
    for (int blk = 0; blk < 3; ++blk) { uW[blk] = (const float*)d_in[i]; ug[blk] = (const float*)d_in[i+1]; ub[blk] = (const float*)d_in[i+2]; i += 3; }
    W1 = (const float*)d_in[i]; fg = (const float*)d_in[i+1]; fb = (const float*)d_in[i+2]; W2 = (const float*)d_in[i+3];
  }

  // ---- workspace carve ----
  char* w = (char*)d_ws;
  auto take = [&](size_t bytes) { void* p = (void*)w; w += (bytes + 255) & ~(size_t)255; return p; };
  int*   n1    = (int*)  take((size_t)NPTS * KNBR * 4);
  int*   n2    = (int*)  take((size_t)NPTS * KNBR * 4);
  int*   nn21  = (int*)  take((size_t)NPTS * 4);
  float* winf1 = (float*)take((size_t)NPTS * KNBR * KP * 4);
  float* winf2 = (float*)take((size_t)NPTS * KNBR * KP * 4);
  float* bufA  = (float*)take((size_t)NPTS * 256 * 4);
  float* bufB  = (float*)take((size_t)NPTS * 256 * 4);
  float* bufC  = (float*)take((size_t)NPTS * 256 * 4);
  float* bufD  = (float*)take((size_t)NPTS * 256 * 4);
  float* bufE  = (float*)take((size_t)NPTS * 256 * 4);   // holds d3
  float* skip0 = (float*)take((size_t)NPTS * 32 * 4);
  float* skip1 = (float*)take((size_t)NPTS * 64 * 4);
  float* skip2 = (float*)take((size_t)NPTS * 128 * 4);
  float* xcat  = (float*)take((size_t)NPTS * 384 * 4);
  float* stats = (float*)take((size_t)2 * 512 * 4);
  float* hbuf  = (float*)take((size_t)NPTS * 64 * 4);

  const int T = 256;
  // ---- geometry (fixed across all layers) ----
  knn16_kernel<<<(NPTS + T - 1) / T, T, 0, stream>>>(pos1, n1);
  knn16_kernel<<<(NPTS + T - 1) / T, T, 0, stream>>>(pos2, n2);
  nn1_kernel  <<<(NPTS + T - 1) / T, T, 0, stream>>>(pos2, pos1, nn21);
  winf_kernel <<<(NPTS * KNBR + T - 1) / T, T, 0, stream>>>(pos1, n1, kp, winf1);
  winf_kernel <<<(NPTS * KNBR + T - 1) / T, T, 0, stream>>>(pos2, n2, kp, winf2);
  initfeat_kernel<<<(NPTS + T - 1) / T, T, 0, stream>>>(pos1, feat1, bufA);
  initfeat_kernel<<<(NPTS + T - 1) / T, T, 0, stream>>>(pos2, feat2, bufC);

  // ---- down blocks ----
  float* x1in = bufA; float* x1out = bufB;
  float* x2in = bufC; float* x2out = bufD;
  float* skips[3] = {skip0, skip1, skip2};
  const int Couts[4] = {32, 64, 128, 256};
  int Cin = 4;
  for (int i = 0; i < 4; ++i) {
    int Cout = Couts[i];
    dim3 g(NPTS / 16, 1);
    kpconv_wmma_kernel<<<g, 128, 0, stream>>>(x1in, n1, winf1, dW[i], x1out, Cin, Cout);
    bn_stats_kernel<<<Cout, T, 0, stream>>>(x1out, stats, Cout);
    bn_apply_kernel<<<(NPTS * Cout + T - 1) / T, T, 0, stream>>>(x1out, stats, dg[i], db[i], 0.1f, Cout);
    kpconv_wmma_kernel<<<g, 128, 0, stream>>>(x2in, n2, winf2, dW[i], x2out, Cin, Cout);
    bn_stats_kernel<<<Cout, T, 0, stream>>>(x2out, stats, Cout);
    bn_apply_kernel<<<(NPTS * Cout + T - 1) / T, T, 0, stream>>>(x2out, stats, dg[i], db[i], 0.1f, Cout);
    float* dst = (i < 3) ? skips[i] : bufE;
    diff_kernel<<<(NPTS * Cout + T - 1) / T, T, 0, stream>>>(x2out, x1out, nn21, dst, Cout);
    float* t = x1in; x1in = x1out; x1out = t;
    t = x2in; x2in = x2out; x2out = t;
    Cin = Cout;
  }

  // ---- up blocks (skip concat, KPConv on cloud-2 geometry) ----
  float* xcur = bufE;                      // width 256
  const int upCx[3] = {256, 128, 64};
  const int upCs[3] = {128, 64, 32};
  const int upCo[3] = {128, 64, 32};
  for (int j = 0; j < 3; ++j) {
    int Cx = upCx[j], Cs = upCs[j], Cc = Cx + Cs, Cout = upCo[j];
    concat_kernel<<<(NPTS * Cc + T - 1) / T, T, 0, stream>>>(xcur, skips[2 - j], xcat, Cx, Cs);
    float* outb = (j & 1) ? bufB : bufA;
    dim3 g(NPTS / 16, 1);
    kpconv_wmma_kernel<<<g, 128, 0, stream>>>(xcat, n2, winf2, uW[j], outb, Cc, Cout);
    bn_stats_kernel<<<Cout, T, 0, stream>>>(outb, stats, Cout);
    bn_apply_kernel<<<(NPTS * Cout + T - 1) / T, T, 0, stream>>>(outb, stats, ug[j], ub[j], 0.1f, Cout);
    xcur = outb;
  }

  // ---- FC head: 32 -> 64 (WMMA) -> BN -> leaky(0.2) -> 64 -> 2 ----
  gemm_wmma_kernel<<<dim3(NPTS / 16, 1), 32, 0, stream>>>(xcur, W1, hbuf, 32, 64);
  bn_stats_kernel<<<64, T, 0, stream>>>(hbuf, stats, 64);
  bn_apply_kernel<<<(NPTS * 64 + T - 1) / T, T, 0, stream>>>(hbuf, stats, fg, fb, 0.2f, 64);
  fc2_kernel<<<(NPTS + T - 1) / T, T, 0, stream>>>(hbuf, W2, (float*)d_out);
}